// Get_Atten_map_mc_84817014161452
// MI455X (gfx1250) — compile-verified
//
#include <hip/hip_runtime.h>
#include <math.h>

typedef __attribute__((ext_vector_type(2))) float v2f;
typedef __attribute__((ext_vector_type(8))) float v8f;

__device__ __forceinline__ float sigf(float x) {
    return 1.0f / (1.0f + __expf(-x));
}

// ---------------------------------------------------------------------------
// Kernel 1: head = min(rel_inds[:, 1:3])
// ---------------------------------------------------------------------------
__global__ void head_kernel(const int* __restrict__ rel, int E, int* __restrict__ head) {
    int e = blockIdx.x * blockDim.x + threadIdx.x;
    if (e < E) {
        int a = rel[3 * e + 1];
        int b = rel[3 * e + 2];
        atomicMin(head, a < b ? a : b);
    }
}

// ---------------------------------------------------------------------------
// Kernel 2: comb[N x 2D] = obj @ [ws_w | wo_w] + [ws_b | wo_b]
// One wave computes one 16x16 tile of C via v_wmma_f32_16x16x4_f32.
// A frag (16x4 f32): lanes 0-15 -> M=lane, {K0,K1}; lanes 16-31 -> M=lane-16, {K2,K3}
// B frag (4x16 f32): mirrored (lanes 0-15 -> N=lane, {K0,K1}; 16-31 -> {K2,K3})
// C (16x16 f32): VGPR r -> M=r (lanes 0-15), M=r+8 (lanes 16-31), N=lane&15
// ---------------------------------------------------------------------------
__global__ void gemm_wmma_kernel(const float* __restrict__ A,
                                 const float* __restrict__ Bs,
                                 const float* __restrict__ Bo,
                                 const float* __restrict__ bs,
                                 const float* __restrict__ bo,
                                 float* __restrict__ C,
                                 int N, int D) {
    const int C2 = 2 * D;
    const int Nt = C2 >> 4;                 // tiles along columns
    const int totalTiles = (N >> 4) * Nt;
    int wv = (blockIdx.x * blockDim.x + threadIdx.x) >> 5;
    if (wv >= totalTiles) return;           // wave-uniform branch: EXEC stays all-1s
    int lane = threadIdx.x & 31;
    int lr = lane & 15;
    int half = lane >> 4;

    int tm = wv / Nt, tn = wv % Nt;
    int m = (tm << 4) + lr;                 // A row for this lane
    int ncol = (tn << 4) + lr;              // global output column

    const float* Bp; const float* biasp; int bc;
    if (ncol < D) { Bp = Bs; biasp = bs; bc = ncol; }
    else          { Bp = Bo; biasp = bo; bc = ncol - D; }

    const float* arow = A + (size_t)m * D + 2 * half;        // lane's K pair base
    const float* bcol = Bp + (size_t)(2 * half) * D + bc;

    v8f c = {};
    for (int k = 0; k < D; k += 4) {
        v2f a, b;
        a.x = arow[k];
        a.y = arow[k + 1];
        b.x = bcol[(size_t)k * D];
        b.y = bcol[(size_t)(k + 1) * D];
        c = __builtin_amdgcn_wmma_f32_16x16x4_f32(
                /*neg_a=*/false, a, /*neg_b=*/false, b,
                /*c_mod=*/(short)0, c, /*reuse_a=*/false, /*reuse_b=*/false);
    }

    float bias = biasp[bc];
    float* crow = C + (size_t)((tm << 4) + 8 * half) * C2 + ncol;
#pragma unroll
    for (int r = 0; r < 8; ++r)
        crow[(size_t)r * C2] = c[r] + bias;
}

// ---------------------------------------------------------------------------
// Kernel 3: per-edge  atten_f = (s[r1] * o[r2] * union[e]) @ w_w + w_b,
// scatter-added into raw[(r1-head)*N + (r2-head)] with f32 global atomics.
// One wave per edge; weight 512x8 transposed into LDS.
// ---------------------------------------------------------------------------
__global__ void edge_kernel(const float* __restrict__ comb,
                            const float* __restrict__ uni,
                            const int* __restrict__ rel,
                            const float* __restrict__ w,    // D x P row-major
                            const float* __restrict__ wb,   // P
                            const int* __restrict__ headp,
                            float* __restrict__ raw,        // N x N x P
                            int N, int D, int P, int E) {
    __shared__ float wT[8 * 512];           // wT[p*D + k]
    int tid = threadIdx.x;
    int DP = D * P;
    for (int idx = tid; idx < DP; idx += blockDim.x) {
        int k = idx / P, p = idx - k * P;
        wT[p * D + k] = w[idx];
    }
    __syncthreads();

    int wv = tid >> 5, lane = tid & 31;
    int e = blockIdx.x * 8 + wv;
    if (e >= E) return;

    int head = *headp;
    int r1 = rel[3 * e + 1];
    int r2 = rel[3 * e + 2];
    const float* srow = comb + (size_t)r1 * (2 * D);
    const float* orow = comb + (size_t)r2 * (2 * D) + D;
    const float* urow = uni + (size_t)e * D;

    float acc[8] = {0.f, 0.f, 0.f, 0.f, 0.f, 0.f, 0.f, 0.f};
    const int steps = D >> 7;               // D / 128
    for (int i = 0; i < steps; ++i) {
        int k0 = 4 * lane + 128 * i;
        float4 s4 = *(const float4*)(srow + k0);
        float4 o4 = *(const float4*)(orow + k0);
        float4 u4 = *(const float4*)(urow + k0);
        float xs[4];
        xs[0] = s4.x * o4.x * u4.x;
        xs[1] = s4.y * o4.y * u4.y;
        xs[2] = s4.z * o4.z * u4.z;
        xs[3] = s4.w * o4.w * u4.w;
#pragma unroll
        for (int j = 0; j < 4; ++j) {
            float x = xs[j];
#pragma unroll
            for (int p = 0; p < 8; ++p)
                acc[p] += x * wT[p * D + k0 + j];
        }
    }

#pragma unroll
    for (int p = 0; p < 8; ++p) {
        float v = acc[p];
#pragma unroll
        for (int off = 16; off; off >>= 1)
            v += __shfl_xor(v, off, 32);
        acc[p] = v;
    }

    if (lane == 0) {
        size_t base = ((size_t)(r1 - head) * N + (size_t)(r2 - head)) * P;
#pragma unroll
        for (int p = 0; p < 8; ++p)
            atomicAdd(&raw[base + p], acc[p] + wb[p]);
    }
}

// ---------------------------------------------------------------------------
// Kernel 4: S[a,p] = sum_b (b==a ? 0 : sigmoid(raw[a,b,p]))   (row sums)
// One wave per row a; coalesced float4 reads.
// ---------------------------------------------------------------------------
__global__ void rowsum_kernel(const float* __restrict__ raw,
                              float* __restrict__ S, int N, int P) {
    int wv = threadIdx.x >> 5, lane = threadIdx.x & 31;
    int a = blockIdx.x * 8 + wv;
    if (a >= N) return;
    const float* row = raw + (size_t)a * N * P;
    float acc[8] = {0.f, 0.f, 0.f, 0.f, 0.f, 0.f, 0.f, 0.f};
    for (int b = lane; b < N; b += 32) {
        if (b == a) continue;
        const float4* p4 = (const float4*)(row + (size_t)b * P);
        float4 v0 = p4[0];
        float4 v1 = p4[1];
        acc[0] += sigf(v0.x); acc[1] += sigf(v0.y);
        acc[2] += sigf(v0.z); acc[3] += sigf(v0.w);
        acc[4] += sigf(v1.x); acc[5] += sigf(v1.y);
        acc[6] += sigf(v1.z); acc[7] += sigf(v1.w);
    }
#pragma unroll
    for (int p = 0; p < 8; ++p) {
        float v = acc[p];
#pragma unroll
        for (int off = 16; off; off >>= 1)
            v += __shfl_xor(v, off, 32);
        acc[p] = v;
    }
    if (lane == 0) {
#pragma unroll
        for (int p = 0; p < 8; ++p)
            S[(size_t)a * P + p] = acc[p];
    }
}

// ---------------------------------------------------------------------------
// Kernel 5: in-place  out[i,j,p] = (i==j ? 0 : sigmoid(raw[i,j,p])) / S[j,p]
// float4 grid-stride; P=8 keeps (i,j) uniform inside each float4.
// ---------------------------------------------------------------------------
__global__ void final_kernel(float* __restrict__ data,
                             const float* __restrict__ S,
                             int N, int P, long total4) {
    int NP = N * P;
    long t = (long)blockIdx.x * blockDim.x + threadIdx.x;
    long stride = (long)gridDim.x * blockDim.x;
    for (; t < total4; t += stride) {
        long base = t * 4;
        int i = (int)(base / NP);
        int r = (int)(base - (long)i * NP);
        int j = r / P;
        float4 o;
        if (i == j) {
            o.x = 0.f; o.y = 0.f; o.z = 0.f; o.w = 0.f;
        } else {
            float4 v = *(const float4*)(data + base);
            float4 s = *(const float4*)(S + (size_t)j * P + (r - j * P));
            o.x = sigf(v.x) / s.x;
            o.y = sigf(v.y) / s.y;
            o.z = sigf(v.z) / s.z;
            o.w = sigf(v.w) / s.w;
        }
        *(float4*)(data + base) = o;
    }
}

// ---------------------------------------------------------------------------
// Host launcher
// ---------------------------------------------------------------------------
extern "C" void kernel_launch(void* const* d_in, const int* in_sizes, int n_in,
                              void* d_out, int out_size, void* d_ws, size_t ws_size,
                              hipStream_t stream) {
    const float* obj  = (const float*)d_in[0];
    const int*   rel  = (const int*)d_in[1];
    const float* uni  = (const float*)d_in[2];
    const float* wsw  = (const float*)d_in[4];
    const float* wsb  = (const float*)d_in[5];
    const float* wow  = (const float*)d_in[6];
    const float* wob  = (const float*)d_in[7];
    const float* ww   = (const float*)d_in[8];
    const float* wbv  = (const float*)d_in[9];
    float* out = (float*)d_out;

    // Derive shapes from flat sizes: ws_w is D*D, obj is N*D, w_w is D*P.
    int D = (int)lround(sqrt((double)in_sizes[4]));
    int N = in_sizes[0] / D;
    int P = in_sizes[8] / D;
    int E = in_sizes[1] / 3;

    // Workspace layout.
    char* wsb8 = (char*)d_ws;
    int*   head_ptr = (int*)wsb8;                         // 4 bytes (padded to 256)
    float* S_ptr    = (float*)(wsb8 + 256);               // N*P floats
    size_t s_bytes  = ((size_t)N * P * sizeof(float) + 255) & ~(size_t)255;
    float* comb     = (float*)(wsb8 + 256 + s_bytes);     // N x 2D floats

    // Zero the dense raw tensor (scatter target) and init head to "infinity".
    hipMemsetAsync(d_out, 0, (size_t)out_size * sizeof(float), stream);
    hipMemsetAsync(head_ptr, 0x7F, sizeof(int), stream);  // 0x7F7F7F7F > any index

    // head = min(rel_inds[:,1:])
    head_kernel<<<(E + 255) / 256, 256, 0, stream>>>(rel, E, head_ptr);

    // comb = obj @ [ws_w | wo_w] + bias   (WMMA f32 16x16x4)
    int totalTiles = (N / 16) * ((2 * D) / 16);
    int gemmBlocks = (totalTiles + 3) / 4;                // 4 waves / 128-thread block
    gemm_wmma_kernel<<<gemmBlocks, 128, 0, stream>>>(obj, wsw, wow, wsb, wob, comb, N, D);

    // Per-edge projection + atomic scatter into raw (= d_out for now).
    edge_kernel<<<(E + 7) / 8, 256, 0, stream>>>(comb, uni, rel, ww, wbv, head_ptr,
                                                 out, N, D, P, E);

    // Row sums of masked sigmoid.
    rowsum_kernel<<<(N + 7) / 8, 256, 0, stream>>>(out, S_ptr, N, P);

    // In-place normalize (broadcast divides by S indexed by the SECOND index j).
    long total4 = ((long)N * N * P) / 4;
    final_kernel<<<8192, 256, 0, stream>>>(out, S_ptr, N, P, total4);
}